// PolicyGradient_tf_36558761624241
// MI455X (gfx1250) — compile-verified
//
#include <hip/hip_runtime.h>

typedef __attribute__((ext_vector_type(2))) float v2f;
typedef __attribute__((ext_vector_type(8))) float v8f;

#define T_STEPS 131072
#define IN_DIM  256
#define HID1    64
#define HID2    32
#define OUT_DIM 8

// ---------------- Phase 0: zero the stats accumulators ----------------
__global__ void zero_stats_kernel(float* stats) {
    stats[threadIdx.x] = 0.0f;   // 512 threads: sum[256] + sumsq[256]
}

// ---------------- Phase 1a: per-column sum / sumsq ----------------
__global__ void stats_kernel(const float* __restrict__ x,
                             float* __restrict__ sum, float* __restrict__ sumsq) {
    const int c = threadIdx.x;                    // 256 threads = 256 columns
    const long r0 = (long)blockIdx.x * 1024;      // 128 blocks x 1024 rows
    float s = 0.0f, s2 = 0.0f;
    for (int r = 0; r < 1024; ++r) {
        float v = x[(r0 + r) * IN_DIM + c];       // consecutive lanes -> consecutive cols
        s += v; s2 += v * v;
    }
    atomicAdd(&sum[c], s);
    atomicAdd(&sumsq[c], s2);
}

// ---------------- Phase 1b: fold BN into W1' and b1' ----------------
__global__ void fold_kernel(const float* __restrict__ sum, const float* __restrict__ sumsq,
                            const float* __restrict__ gamma, const float* __restrict__ beta,
                            const float* __restrict__ W1, const float* __restrict__ b1,
                            float* __restrict__ W1f, float* __restrict__ b1f) {
    __shared__ float s_shift[IN_DIM];
    const int c = threadIdx.x;                    // 256 threads
    float mean = sum[c]   * (1.0f / T_STEPS);
    float var  = sumsq[c] * (1.0f / T_STEPS) - mean * mean;  // biased, as keras BN
    float sc = gamma[c] * rsqrtf(var + 1e-3f);
    float sh = beta[c] - mean * sc;
    s_shift[c] = sh;
    __syncthreads();
    for (int j = 0; j < HID1; ++j)
        W1f[c * HID1 + j] = sc * W1[c * HID1 + j];
    if (c < HID1) {
        float acc = b1[c];
        for (int cc = 0; cc < IN_DIM; ++cc)
            acc = fmaf(s_shift[cc], W1[cc * HID1 + c], acc);
        b1f[c] = acc;
    }
}

// ---------------- Phase 2: A = x @ W1f + b1f via V_WMMA_F32_16X16X4_F32 ----------------
// 128 threads = 4 waves; wave w owns a 16-row stripe; 4 N-tiles of 16; K-loop of 64.
__global__ void gemm_wmma_kernel(const float* __restrict__ x, const float* __restrict__ W1f,
                                 const float* __restrict__ b1f, float* __restrict__ Abuf) {
    const int wave  = threadIdx.x >> 5;
    const int lane  = threadIdx.x & 31;
    const int lhalf = lane >> 4;      // 0: K=0,1 / M rows 0..7 of C ; 1: K=2,3 / M rows 8..15
    const int lmod  = lane & 15;
    const long row0 = (long)blockIdx.x * 64 + wave * 16;

    v8f acc[4];
#pragma unroll
    for (int n = 0; n < 4; ++n) {
        float bv = b1f[n * 16 + lmod];            // C[M,N] = b1'[N], independent of M
        acc[n] = (v8f){bv, bv, bv, bv, bv, bv, bv, bv};
    }

    const float* arow = x + (row0 + lmod) * IN_DIM + lhalf * 2;
#pragma unroll 4
    for (int kk = 0; kk < 64; ++kk) {
        const int kb = kk * 4 + lhalf * 2;
        v2f a;                                    // A 16x4 f32: lane<16 K=0,1 ; lane>=16 K=2,3
        a.x = arow[kk * 4 + 0];
        a.y = arow[kk * 4 + 1];
#pragma unroll
        for (int n = 0; n < 4; ++n) {
            v2f b;                                // B 4x16 f32, mirrored lane layout
            b.x = W1f[(kb + 0) * HID1 + n * 16 + lmod];
            b.y = W1f[(kb + 1) * HID1 + n * 16 + lmod];
            acc[n] = __builtin_amdgcn_wmma_f32_16x16x4_f32(
                false, a, false, b, (short)0, acc[n], false, false);
        }
    }

#pragma unroll
    for (int n = 0; n < 4; ++n) {
        const int col = n * 16 + lmod;
#pragma unroll
        for (int r = 0; r < 8; ++r) {
            const long row = row0 + r + 8 * lhalf;   // C/D: VGPR r -> M=r (+8 for upper half)
            Abuf[row * HID1 + col] = acc[n][r];
        }
    }
}

// ---------------- Phase 3: sequential scan, single wave32 ----------------
__global__ void scan_kernel(const float* __restrict__ Abuf, const float* __restrict__ W1,
                            const float* __restrict__ W2, const float* __restrict__ b2,
                            const float* __restrict__ W3, const float* __restrict__ b3,
                            float* __restrict__ out) {
    __shared__ __align__(16) float lds_h1[HID1];
    __shared__ __align__(16) float lds_h2[HID2];
    __shared__ __align__(16) float lds_o[OUT_DIM];

    const int lane = threadIdx.x;                 // 0..31, one wave
    // W1 tail (rows 256..263): lane holds columns 2*lane, 2*lane+1
    float w1p0[OUT_DIM], w1p1[OUT_DIM];
#pragma unroll
    for (int m = 0; m < OUT_DIM; ++m) {
        w1p0[m] = W1[(IN_DIM + m) * HID1 + 2 * lane + 0];
        w1p1[m] = W1[(IN_DIM + m) * HID1 + 2 * lane + 1];
    }
    // W2 column per lane (64 VGPRs)
    float w2c[HID1];
#pragma unroll
    for (int j = 0; j < HID1; ++j) w2c[j] = W2[j * HID2 + lane];
    // W3 column: groups of 8 lanes compute o[lane&7] redundantly
    const int m8 = lane & 7;
    float w3c[HID2];
#pragma unroll
    for (int k = 0; k < HID2; ++k) w3c[k] = W3[k * OUT_DIM + m8];
    const float b2v = b2[lane];
    const float b3v = b3[m8];

    float h[OUT_DIM];
#pragma unroll
    for (int m = 0; m < OUT_DIM; ++m) h[m] = 0.0f;

    for (int t = 0; t < T_STEPS; ++t) {
        if (t + 16 < T_STEPS)
            __builtin_prefetch(&Abuf[(long)(t + 16) * HID1], 0, 0);

        // u = A[t] + h @ W1tail  (lane -> h1 cols 2*lane, 2*lane+1)
        float u0 = Abuf[(long)t * HID1 + 2 * lane + 0];
        float u1 = Abuf[(long)t * HID1 + 2 * lane + 1];
#pragma unroll
        for (int m = 0; m < OUT_DIM; ++m) {
            u0 = fmaf(h[m], w1p0[m], u0);
            u1 = fmaf(h[m], w1p1[m], u1);
        }
        lds_h1[2 * lane + 0] = fmaxf(u0, 0.0f);
        lds_h1[2 * lane + 1] = fmaxf(u1, 0.0f);
        __syncthreads();

        // h2[lane] = relu(b2 + h1 @ W2[:,lane]) ; uniform-address b128 LDS broadcasts
        float acc = b2v;
#pragma unroll
        for (int j4 = 0; j4 < HID1 / 4; ++j4) {
            float4 hv = ((const float4*)lds_h1)[j4];
            acc = fmaf(hv.x, w2c[4 * j4 + 0], acc);
            acc = fmaf(hv.y, w2c[4 * j4 + 1], acc);
            acc = fmaf(hv.z, w2c[4 * j4 + 2], acc);
            acc = fmaf(hv.w, w2c[4 * j4 + 3], acc);
        }
        lds_h2[lane] = fmaxf(acc, 0.0f);
        __syncthreads();

        // o[m8] = b3 + h2 @ W3[:,m8]
        float o = b3v;
#pragma unroll
        for (int k4 = 0; k4 < HID2 / 4; ++k4) {
            float4 hv = ((const float4*)lds_h2)[k4];
            o = fmaf(hv.x, w3c[4 * k4 + 0], o);
            o = fmaf(hv.y, w3c[4 * k4 + 1], o);
            o = fmaf(hv.z, w3c[4 * k4 + 2], o);
            o = fmaf(hv.w, w3c[4 * k4 + 3], o);
        }
        if (lane < OUT_DIM) {
            lds_o[lane] = o;
            out[(long)t * OUT_DIM + lane] = o;
        }
        __syncthreads();

        float4 o0 = ((const float4*)lds_o)[0];
        float4 o1 = ((const float4*)lds_o)[1];
        h[0] = o0.x; h[1] = o0.y; h[2] = o0.z; h[3] = o0.w;
        h[4] = o1.x; h[5] = o1.y; h[6] = o1.z; h[7] = o1.w;
        __syncthreads();
    }
}

extern "C" void kernel_launch(void* const* d_in, const int* in_sizes, int n_in,
                              void* d_out, int out_size, void* d_ws, size_t ws_size,
                              hipStream_t stream) {
    const float* x     = (const float*)d_in[0];
    const float* gamma = (const float*)d_in[1];
    const float* beta  = (const float*)d_in[2];
    const float* W1    = (const float*)d_in[3];
    const float* b1    = (const float*)d_in[4];
    const float* W2    = (const float*)d_in[5];
    const float* b2    = (const float*)d_in[6];
    const float* W3    = (const float*)d_in[7];
    const float* b3    = (const float*)d_in[8];
    float* out = (float*)d_out;

    // workspace layout (floats)
    float* wsf   = (float*)d_ws;
    float* sum   = wsf + 0;        // 256
    float* sumsq = wsf + 256;      // 256
    float* b1f   = wsf + 512;      // 64
    float* W1f   = wsf + 576;      // 256*64 = 16384
    float* Abuf  = wsf + 16960;    // T*64 (16-byte aligned offset)

    zero_stats_kernel<<<1, 512, 0, stream>>>(sum);
    stats_kernel<<<T_STEPS / 1024, 256, 0, stream>>>(x, sum, sumsq);
    fold_kernel<<<1, 256, 0, stream>>>(sum, sumsq, gamma, beta, W1, b1, W1f, b1f);
    gemm_wmma_kernel<<<T_STEPS / 64, 128, 0, stream>>>(x, W1f, b1f, Abuf);
    scan_kernel<<<1, 32, 0, stream>>>(Abuf, W1, W2, b2, W3, b3, out);
}